// RelationalBertSelfAttention_412316860778
// MI455X (gfx1250) — compile-verified
//
#include <hip/hip_runtime.h>

// ---------------------------------------------------------------------------
// RelationalBertSelfAttention for MI455X (gfx1250, wave32, WMMA)
// B=4, S=512, D=768, H=12, DH=64.
//
// Kernel 1: QKV projection, v_wmma_f32_16x16x4_f32, -> ws [3][B,H,S,DH]
// Kernel 2: fused flash attention with k-split. One WG = (b, 16-query tile,
//   k-range). 16 waves, dual role:
//     wave w (all 16)  = "query-wave w": relational score & relational context
//                        as cross-head WMMAs against the LDS R tile
//     wave w (w < 12)  = "head-wave w": content scores, online softmax, P@V
//   dep_rel tile staged in LDS via GLOBAL_LOAD_ASYNC_TO_LDS (double-buffered),
//   reused by all 12 heads (12x HBM traffic reduction on the 256MB tensor).
// Kernel 3: merge k-split partials (m,l,acc) and write [B,S,D].
// ---------------------------------------------------------------------------

#define B_   4
#define S_   512
#define D_   768
#define H_   12
#define DH_  64
#define SCALE_ 0.125f           // 1/sqrt(64)
#define QT_  (S_ / 16)          // 32 query tiles
#define KS_  4                  // k-splits
#define KLEN_ (S_ / KS_)        // 128 keys per split
#define PART_STRIDE_ 1056       // 16x64 acc + 16 m + 16 l

#if defined(__has_builtin)
#if __has_builtin(__builtin_amdgcn_global_load_async_to_lds_b128) && \
    __has_builtin(__builtin_amdgcn_s_wait_asynccnt)
#define HAVE_ASYNC_LDS 1
#endif
#endif
#ifndef HAVE_ASYNC_LDS
#define HAVE_ASYNC_LDS 0
#endif

typedef float v2f __attribute__((ext_vector_type(2)));
typedef float v8f __attribute__((ext_vector_type(8)));

#if HAVE_ASYNC_LDS
typedef int v4i_ __attribute__((vector_size(16)));
typedef __attribute__((address_space(1))) v4i_ gv4i;   // global 16B vector
typedef __attribute__((address_space(3))) v4i_ lv4i;   // LDS 16B vector
#endif

static __device__ __forceinline__ v8f wmma_f32(v2f a, v2f b, v8f c) {
    // D = A(16x4 f32) * B(4x16 f32) + C(16x16 f32)
    return __builtin_amdgcn_wmma_f32_16x16x4_f32(
        false, a, false, b, (short)0, c, false, false);
}

// Issue a 16x16x64-f32 R-tile copy global -> LDS (64KB), 8 x b128 per thread.
static __device__ __forceinline__ void issue_tile_copy(
    float* dst, const float* __restrict__ Rb, int q0, int k0, int tid)
{
    for (int i = tid * 4; i < 16 * 16 * DH_; i += 512 * 4) {
        const int qq  = i >> 10;
        const int rem = i & 1023;
        const int kk  = rem >> 6;
        const int d   = rem & 63;
        const float* src = Rb + (((size_t)(q0 + qq) * S_) + k0 + kk) * DH_ + d;
#if HAVE_ASYNC_LDS
        __builtin_amdgcn_global_load_async_to_lds_b128(
            (gv4i*)src, (lv4i*)(dst + i), 0, 0);
#else
        *(float4*)(dst + i) = *(const float4*)src;
#endif
    }
}

static __device__ __forceinline__ void wait_tile_copy() {
#if HAVE_ASYNC_LDS
    __builtin_amdgcn_s_wait_asynccnt(0);
#endif
}

// ---------------------------------------------------------------------------
// Kernel 1: out[m, n] = sum_k hidden[m, k] * W[n, k] + bias[n]
// A-frag (16x4): lane L -> row M=L%16, K=(L/16)*2+{0,1}
// B-frag (4x16): lane L -> col N=L%16, K=(L/16)*2+{0,1}
// C layout:      vgpr r, lane L -> M=r+8*(L/16), N=L%16
// ---------------------------------------------------------------------------
__global__ __launch_bounds__(32) void qkv_kernel(
    const float* __restrict__ hidden,
    const float* __restrict__ Wq, const float* __restrict__ bq,
    const float* __restrict__ Wk, const float* __restrict__ bk,
    const float* __restrict__ Wv, const float* __restrict__ bv,
    float* __restrict__ qkv_ws)
{
    const int mt  = blockIdx.x;   // B*S/16
    const int nt  = blockIdx.y;   // D/16
    const int mat = blockIdx.z;   // 0=Q 1=K 2=V

    const float* W    = (mat == 0) ? Wq : (mat == 1) ? Wk : Wv;
    const float* bias = (mat == 0) ? bq : (mat == 1) ? bk : bv;
    float* out = qkv_ws + (size_t)mat * ((size_t)B_ * H_ * S_ * DH_);

    const int lane = threadIdx.x & 31;
    const int lo = lane & 15;
    const int hi = lane >> 4;
    const int kbase = hi * 2;

    const float* arow = hidden + (size_t)(mt * 16 + lo) * D_;
    const float* brow = W      + (size_t)(nt * 16 + lo) * D_;

    v8f c = {};
    for (int k = 0; k < D_; k += 4) {
        v2f a = *(const v2f*)(arow + k + kbase);
        v2f b = *(const v2f*)(brow + k + kbase);
        c = wmma_f32(a, b, c);
    }

    const int nn = nt * 16 + lo;
    const int h  = nn >> 6;
    const int dh = nn & 63;
    const float bb = bias[nn];

#pragma unroll
    for (int r = 0; r < 8; ++r) {
        const int mm   = mt * 16 + r + 8 * hi;
        const int bidx = mm / S_;
        const int s    = mm % S_;
        out[(((size_t)bidx * H_ + h) * S_ + s) * DH_ + dh] = c[r] + bb;
    }
}

// ---------------------------------------------------------------------------
// Kernel 2: fused attention, all matmuls on WMMA, async double-buffered R.
// grid = (QT, B, KS); block = 16 waves (512 threads).
// LDS: Rs0/Rs1[16q][16k][64d] 2x64KB | Srel[12h][16q][16k] 12KB
//      Ps[12h][16q][16k] 12KB | alpha[16h][16q] 1KB   (total ~156.7KB)
// ---------------------------------------------------------------------------
__global__ __launch_bounds__(512) void attn_kernel(
    const float* __restrict__ qkv_ws,   // [3][B][H][S][DH]
    const float* __restrict__ rel,      // [B][S][S][DH]
    const float* __restrict__ mask,     // [B][S]
    float* __restrict__ part)           // [B][H][QT][KS][PART_STRIDE]
{
    extern __shared__ float smem[];
    float* Rs0  = smem;                 // 16384
    float* Rs1  = smem + 16384;         // 16384
    float* Srel = smem + 32768;         // 12*256 = 3072
    float* Ps   = smem + 35840;         // 12*256 = 3072
    float* Alds = smem + 38912;         // 256

    const int qt  = blockIdx.x;
    const int b   = blockIdx.y;
    const int ks  = blockIdx.z;
    const int tid = threadIdx.x;
    const int wid = tid >> 5;          // wave id: query role (0..15), head role (<12)
    const int lane = tid & 31;
    const int lo   = lane & 15;
    const int hi   = lane >> 4;
    const int q0   = qt * 16;

    const size_t mat_sz = (size_t)B_ * H_ * S_ * DH_;
    const float* Qg = qkv_ws;
    const float* Kh = qkv_ws + mat_sz     + ((size_t)b * H_ + wid) * S_ * DH_;
    const float* Vh = qkv_ws + 2 * mat_sz + ((size_t)b * H_ + wid) * S_ * DH_;
    const float* Rb = rel + (size_t)b * S_ * S_ * DH_;
    const float* mb = mask + (size_t)b * S_;

    // init padding-head alpha rows (heads 12..15 never write these)
    for (int i = tid; i < 256; i += 512) Alds[i] = 1.0f;

    // ---- head-role A-fragments: Q(16q x 64d) for head `wid` ----
    v2f qa[16];
    if (wid < H_) {
#pragma unroll
        for (int st = 0; st < 16; ++st) {
            const float* qp = Qg + (((size_t)b * H_ + wid) * S_ + q0 + lo) * DH_
                                 + st * 4 + hi * 2;
            qa[st].x = qp[0]; qa[st].y = qp[1];
        }
    }
    // ---- query-role A-fragments: Qstack(16h x 64d) for query q0+wid ----
    v2f qa2[16];
#pragma unroll
    for (int st = 0; st < 16; ++st) {
        if (lo < H_) {
            const float* qp = Qg + (((size_t)b * H_ + lo) * S_ + q0 + wid) * DH_
                                 + st * 4 + hi * 2;
            qa2[st].x = qp[0]; qa2[st].y = qp[1];
        } else {
            qa2[st].x = 0.0f; qa2[st].y = 0.0f;
        }
    }

    v8f acc[4], accrel[4];
#pragma unroll
    for (int t = 0; t < 4; ++t) { acc[t] = (v8f){}; accrel[t] = (v8f){}; }
    float mrow[8], lrow[8];
#pragma unroll
    for (int r = 0; r < 8; ++r) { mrow[r] = -1.0e30f; lrow[r] = 0.0f; }

    const int kbeg = ks * KLEN_;
    const int kend = kbeg + KLEN_;

    // prologue: stage first R tile
    float* Rcur = Rs0;
    float* Rnxt = Rs1;
    issue_tile_copy(Rcur, Rb, q0, kbeg, tid);
    wait_tile_copy();
    __syncthreads();

    for (int k0 = kbeg; k0 < kend; k0 += 16) {
        // kick off next tile into the other buffer (overlaps all compute below)
        if (k0 + 16 < kend)
            issue_tile_copy(Rnxt, Rb, q0, k0 + 16, tid);

        // ------- query role: S_rel(16h x 16k) = Qstack @ R[q]^T -------
        {
            v8f sr = {};
#pragma unroll
            for (int st = 0; st < 16; ++st) {
                v2f bf = *(const v2f*)(Rcur + wid * 1024 + lo * 64 + st * 4 + hi * 2);
                sr = wmma_f32(qa2[st], bf, sr);
            }
#pragma unroll
            for (int r = 0; r < 8; ++r) {
                const int row = r + 8 * hi;
                if (row < H_) Srel[row * 256 + wid * 16 + lo] = sr[r];
            }
        }
        __syncthreads();

        // ------- head role: content scores + softmax + P, alpha -------
        if (wid < H_) {
            v8f s = {};
#pragma unroll
            for (int st = 0; st < 16; ++st) {
                const float* kp = Kh + (size_t)(k0 + lo) * DH_ + st * 4 + hi * 2;
                v2f bf; bf.x = kp[0]; bf.y = kp[1];
                s = wmma_f32(qa[st], bf, s);
            }
            const float mk = mb[k0 + lo];
#pragma unroll
            for (int r = 0; r < 8; ++r) {
                s[r] = (s[r] + Srel[wid * 256 + (r + 8 * hi) * 16 + lo]) * SCALE_ + mk;
                float v = s[r];
#pragma unroll
                for (int off = 1; off < 16; off <<= 1)
                    v = fmaxf(v, __shfl_xor(v, off, 32));
                const float mnew  = fmaxf(mrow[r], v);
                const float alpha = __expf(mrow[r] - mnew);
                float p  = __expf(s[r] - mnew);
                float ps = p;
#pragma unroll
                for (int off = 1; off < 16; off <<= 1)
                    ps += __shfl_xor(ps, off, 32);
                lrow[r] = lrow[r] * alpha + ps;
                mrow[r] = mnew;
#pragma unroll
                for (int t = 0; t < 4; ++t) acc[t][r] *= alpha;
                Ps[wid * 256 + (r + 8 * hi) * 16 + lo] = p;
                if (lo == 0) Alds[wid * 16 + (r + 8 * hi)] = alpha;
            }
        }
        __syncthreads();

        // ------------- head role: ctx_con += P @ V -------------
        if (wid < H_) {
            v2f pa[4];
#pragma unroll
            for (int st = 0; st < 4; ++st)
                pa[st] = *(const v2f*)(Ps + wid * 256 + lo * 16 + st * 4 + hi * 2);
#pragma unroll
            for (int t = 0; t < 4; ++t) {
#pragma unroll
                for (int st = 0; st < 4; ++st) {
                    const float* vp = Vh + (size_t)(k0 + st * 4 + hi * 2) * DH_
                                         + t * 16 + lo;
                    v2f bf; bf.x = vp[0]; bf.y = vp[DH_];
                    acc[t] = wmma_f32(pa[st], bf, acc[t]);
                }
            }
        }

        // ---- query role: ctx_rel = ctx_rel*alpha[h] + P^T(16h x 16k) @ R[q] ----
        {
            float ar[8];
#pragma unroll
            for (int r = 0; r < 8; ++r)
                ar[r] = Alds[(r + 8 * hi) * 16 + wid];
#pragma unroll
            for (int t = 0; t < 4; ++t)
#pragma unroll
                for (int r = 0; r < 8; ++r)
                    accrel[t][r] *= ar[r];

            v2f pra[4];
#pragma unroll
            for (int st = 0; st < 4; ++st)
                pra[st] = *(const v2f*)(Ps + lo * 256 + wid * 16 + st * 4 + hi * 2);
#pragma unroll
            for (int t = 0; t < 4; ++t) {
#pragma unroll
                for (int st = 0; st < 4; ++st) {
                    const float* rp = Rcur + wid * 1024 + (st * 4 + hi * 2) * 64
                                           + t * 16 + lo;
                    v2f bf; bf.x = rp[0]; bf.y = rp[64];
                    accrel[t] = wmma_f32(pra[st], bf, accrel[t]);
                }
            }
        }

        // next tile is ready only after every wave's async ops completed
        wait_tile_copy();
        __syncthreads();
        float* tswap = Rcur; Rcur = Rnxt; Rnxt = tswap;
    }

    // ---------------- epilogue: combine roles, emit partial ----------------
#pragma unroll
    for (int t = 0; t < 4; ++t)      // query role: ctxrel[h][q][d] -> Rs0
#pragma unroll
        for (int r = 0; r < 8; ++r)
            Rs0[(r + 8 * hi) * 1024 + wid * 64 + t * 16 + lo] = accrel[t][r];
    __syncthreads();

    if (wid < H_) {                  // head role: add & store unnormalized partial
        float* pb = part + (((size_t)(b * H_ + wid) * QT_ + qt) * KS_ + ks) * PART_STRIDE_;
#pragma unroll
        for (int t = 0; t < 4; ++t) {
#pragma unroll
            for (int r = 0; r < 8; ++r) {
                const int row = r + 8 * hi;
                pb[row * 64 + t * 16 + lo] =
                    acc[t][r] + Rs0[wid * 1024 + row * 64 + t * 16 + lo];
            }
        }
        if (lo == 0) {
#pragma unroll
            for (int r = 0; r < 8; ++r) {
                pb[1024 + r + 8 * hi] = mrow[r];
                pb[1040 + r + 8 * hi] = lrow[r];
            }
        }
    }
}

// ---------------------------------------------------------------------------
// Kernel 3: merge KS_ flash partials, normalize, write [B][S][H*DH].
// ---------------------------------------------------------------------------
__global__ __launch_bounds__(256) void reduce_kernel(
    const float* __restrict__ part, float* __restrict__ out)
{
    const int qt = blockIdx.x, h = blockIdx.y, b = blockIdx.z;
    const float* p0 = part + (((size_t)(b * H_ + h) * QT_ + qt) * KS_) * PART_STRIDE_;

    for (int e = threadIdx.x; e < 1024; e += 256) {
        const int row = e >> 6, d = e & 63;
        float m = -1.0e30f;
#pragma unroll
        for (int k = 0; k < KS_; ++k)
            m = fmaxf(m, p0[k * PART_STRIDE_ + 1024 + row]);
        float acc = 0.0f, l = 0.0f;
#pragma unroll
        for (int k = 0; k < KS_; ++k) {
            const float f = __expf(p0[k * PART_STRIDE_ + 1024 + row] - m);
            l   += p0[k * PART_STRIDE_ + 1040 + row] * f;
            acc += p0[k * PART_STRIDE_ + e] * f;
        }
        out[((size_t)b * S_ + qt * 16 + row) * D_ + h * DH_ + d] = acc / l;
    }
}

// ---------------------------------------------------------------------------
extern "C" void kernel_launch(void* const* d_in, const int* in_sizes, int n_in,
                              void* d_out, int out_size, void* d_ws, size_t ws_size,
                              hipStream_t stream) {
    const float* hidden = (const float*)d_in[0];
    const float* amask  = (const float*)d_in[1];
    const float* rel    = (const float*)d_in[2];
    const float* Wq     = (const float*)d_in[3];
    const float* bq     = (const float*)d_in[4];
    const float* Wk     = (const float*)d_in[5];
    const float* bk     = (const float*)d_in[6];
    const float* Wv     = (const float*)d_in[7];
    const float* bv     = (const float*)d_in[8];
    float* out = (float*)d_out;

    float* qkv  = (float*)d_ws;                              // 3*B*H*S*DH floats
    float* part = qkv + (size_t)3 * B_ * H_ * S_ * DH_;      // B*H*QT*KS*1056 floats

    dim3 gA(B_ * S_ / 16, D_ / 16, 3);
    qkv_kernel<<<gA, 32, 0, stream>>>(hidden, Wq, bq, Wk, bk, Wv, bv, qkv);

    const size_t smem = (size_t)(2 * 16384 + 3072 + 3072 + 256) * sizeof(float);
    (void)hipFuncSetAttribute((const void*)attn_kernel,
                              hipFuncAttributeMaxDynamicSharedMemorySize, (int)smem);
    dim3 gB(QT_, B_, KS_);
    attn_kernel<<<gB, 512, smem, stream>>>(qkv, rel, amask, part);

    dim3 gC(QT_, H_, B_);
    reduce_kernel<<<gC, 256, 0, stream>>>(part, out);
}